// ripple_carry_adder_13786845020438
// MI455X (gfx1250) — compile-verified
//
#include <hip/hip_runtime.h>
#include <cstdint>

typedef __attribute__((ext_vector_type(4))) unsigned int v4u;
typedef __attribute__((ext_vector_type(8))) int v8i;
typedef __attribute__((ext_vector_type(4))) int v4i;

#define N_ROWS          2097152
#define N_BITS          16
#define ROWS_PER_TILE   256
#define TILE_FLOATS     (ROWS_PER_TILE * N_BITS)     // 4096 floats
#define TILE_BYTES      (TILE_FLOATS * 4)            // 16384 bytes
#define TILE_E8         (TILE_BYTES / 8)             // 2048 8-byte elements
#define NUM_BLOCKS      1024
#define TILES_TOTAL     (N_ROWS / ROWS_PER_TILE)     // 8192
#define TILES_PER_BLOCK (TILES_TOTAL / NUM_BLOCKS)   // 8

// ---- TDM descriptor helpers (CDNA5 ISA) ----

__device__ __forceinline__ v4u tdm_g0(uint32_t lds_addr, uint64_t gaddr) {
  v4u g0;
  g0.x = 1u;                                   // count=1, user descriptor
  g0.y = lds_addr;                             // LDS byte address
  g0.z = (uint32_t)gaddr;                      // global_addr[31:0]
  g0.w = (uint32_t)((gaddr >> 32) & 0x1FFFFFFull) | (2u << 30); // addr[56:32], type=2
  return g0;
}

__device__ __forceinline__ v8i tdm_g1_contig16k() {
  // 1-D contiguous 16KB transfer: data_size=8B, tile_dim0=tensor_dim0=2048
  v8i g1;
  g1[0] = (int)(3u << 16);                         // wg_mask=0, data_size=3 (8B)
  g1[1] = (int)((TILE_E8 & 0xFFFFu) << 16);        // tensor_dim0[15:0] @bits63:48
  g1[2] = (int)((TILE_E8 >> 16) | (1u << 16));     // tensor_dim0[31:16], tensor_dim1=1
  g1[3] = (int)((TILE_E8 & 0xFFFFu) << 16);        // tensor_dim1 hi=0, tile_dim0=2048
  g1[4] = 1;                                       // tile_dim1=1, tile_dim2=0 (unused)
  g1[5] = (int)TILE_E8;                            // tensor_dim0_stride[31:0]
  g1[6] = 0;                                       // stride0 hi, stride1 lo
  g1[7] = 0;                                       // stride1 hi
  return g1;
}

__device__ __forceinline__ void tdm_load(v4u g0, v8i g1, v4i g2, v4i g3) {
#if defined(__clang_major__) && (__clang_major__ >= 23)
  v8i z = (v8i){0, 0, 0, 0, 0, 0, 0, 0};
  __builtin_amdgcn_tensor_load_to_lds(g0, g1, g2, g3, z, 0);
#else
  __builtin_amdgcn_tensor_load_to_lds(g0, g1, g2, g3, 0);
#endif
}

__device__ __forceinline__ void tdm_store(v4u g0, v8i g1, v4i g2, v4i g3) {
#if defined(__clang_major__) && (__clang_major__ >= 23)
  v8i z = (v8i){0, 0, 0, 0, 0, 0, 0, 0};
  __builtin_amdgcn_tensor_store_from_lds(g0, g1, g2, g3, z, 0);
#else
  __builtin_amdgcn_tensor_store_from_lds(g0, g1, g2, g3, 0);
#endif
}

__device__ __forceinline__ uint32_t lds_off(const void* p) {
  // Flat LDS addresses carry the aperture in the high 32 bits; low 32 bits are
  // the byte offset within the workgroup's LDS allocation (ISA §10.2).
  return (uint32_t)(uintptr_t)p;
}

// ---- Kernel ----

__global__ __launch_bounds__(256) void ripple_carry_tdm_kernel(
    const float* __restrict__ x, const float* __restrict__ y,
    float* __restrict__ out) {
  __shared__ float sx[2][TILE_FLOATS];
  __shared__ float sy[2][TILE_FLOATS];
  __shared__ float ss[2][TILE_FLOATS];
  __shared__ float sc[2][TILE_FLOATS];

  const int t = threadIdx.x;
  const int wave = t >> 5;
  const int tile0 = blockIdx.x * TILES_PER_BLOCK;

  const uint64_t xg = (uint64_t)(uintptr_t)x;
  const uint64_t yg = (uint64_t)(uintptr_t)y;
  const uint64_t sg = (uint64_t)(uintptr_t)out;
  const uint64_t cg = sg + (uint64_t)N_ROWS * N_BITS * 4ull;

  const v8i g1 = tdm_g1_contig16k();
  const v4i gz = (v4i){0, 0, 0, 0};

  uint32_t o_sx[2] = {lds_off(&sx[0][0]), lds_off(&sx[1][0])};
  uint32_t o_sy[2] = {lds_off(&sy[0][0]), lds_off(&sy[1][0])};
  uint32_t o_ss[2] = {lds_off(&ss[0][0]), lds_off(&ss[1][0])};
  uint32_t o_sc[2] = {lds_off(&sc[0][0]), lds_off(&sc[1][0])};

  // Prologue: DMA tile0 inputs into buffer 0 (one issue per block, by wave 0).
  if (wave == 0) {
    tdm_load(tdm_g0(o_sx[0], xg + (uint64_t)tile0 * TILE_BYTES), g1, gz, gz);
    tdm_load(tdm_g0(o_sy[0], yg + (uint64_t)tile0 * TILE_BYTES), g1, gz, gz);
  }

  for (int it = 0; it < TILES_PER_BLOCK; ++it) {
    const int tile = tile0 + it;
    const int buf = it & 1;
    const int nbuf = buf ^ 1;

    if (wave == 0) {
      const bool have_next = (it + 1) < TILES_PER_BLOCK;
      if (have_next) {
        tdm_load(tdm_g0(o_sx[nbuf], xg + (uint64_t)(tile + 1) * TILE_BYTES), g1, gz, gz);
        tdm_load(tdm_g0(o_sy[nbuf], yg + (uint64_t)(tile + 1) * TILE_BYTES), g1, gz, gz);
      }
      // TENSORcnt completes in issue order. Ops issued after L(it) are
      // S(it-1) (2 ops, if it>0) and L(it+1) (2 ops, if issued). Waiting down
      // to that count guarantees L(it) -- and every prior store -- is done.
      if (it > 0 && have_next)       __builtin_amdgcn_s_wait_tensorcnt(4);
      else if (it > 0 || have_next)  __builtin_amdgcn_s_wait_tensorcnt(2);
      else                           __builtin_amdgcn_s_wait_tensorcnt(0);
    }
    __syncthreads();  // tile `it` inputs visible in LDS to all waves

    // One 16-bit ripple-carry row per thread, straight from/to LDS.
    {
      const float* xr = &sx[buf][t * N_BITS];
      const float* yr = &sy[buf][t * N_BITS];
      float* sr = &ss[buf][t * N_BITS];
      float* cr = &sc[buf][t * N_BITS];
      float c = 0.0f;
#pragma unroll
      for (int i = 0; i < N_BITS; ++i) {
        const float a = xr[i];
        const float b = yr[i];
        const float s1 = a + b - 2.0f * a * b;   // xor(a,b)
        const float c1 = a * b;                  // and(a,b)
        const float s  = s1 + c - 2.0f * s1 * c; // xor(s1,c_in)
        const float c2 = s1 * c;                 // and(s1,c_in)
        c = c1 + c2 - c1 * c2;                   // or(c1,c2)
        sr[i] = s;
        cr[i] = c;
      }
    }
    __syncthreads();  // tile `it` outputs staged in LDS

    if (wave == 0) {
      tdm_store(tdm_g0(o_ss[buf], sg + (uint64_t)tile * TILE_BYTES), g1, gz, gz);
      tdm_store(tdm_g0(o_sc[buf], cg + (uint64_t)tile * TILE_BYTES), g1, gz, gz);
    }
    // LDS reuse is safe: in-order TENSORcnt + the wait above ensure S(it-2)
    // finished before compute overwrites ss/sc[buf], and L(it+1) targets the
    // opposite input buffer whose consumers passed the barrier last iteration.
  }
  // S_ENDPGM performs an implicit wait-idle, flushing the final tensor stores.
}

extern "C" void kernel_launch(void* const* d_in, const int* in_sizes, int n_in,
                              void* d_out, int out_size, void* d_ws, size_t ws_size,
                              hipStream_t stream) {
  const float* x = (const float*)d_in[0];
  const float* y = (const float*)d_in[1];
  float* out = (float*)d_out;
  ripple_carry_tdm_kernel<<<NUM_BLOCKS, 256, 0, stream>>>(x, y, out);
}